// RBFFeatureInterpolator_90383291777516
// MI455X (gfx1250) — compile-verified
//
#include <hip/hip_runtime.h>
#include <hip/hip_bf16.h>
#include <math.h>

typedef __attribute__((ext_vector_type(2))) float v2f;
typedef __attribute__((ext_vector_type(8))) float v8f;
typedef unsigned long long u64;
typedef unsigned int u32;

constexpr int kB  = 4;
constexpr int kNQ = 8192;
constexpr int kNS = 4096;
constexpr int kF  = 256;
constexpr int kTiles  = kNS / 16;   // 256 sensor tiles of 16
constexpr int kQTperB = kNQ / 16;   // 512 query tiles per batch

// ---------------------------------------------------------------------------
// Prep: pack sensor coords for the WMMA A-operand.
//   P01[s] = (-2x, -2y)      -> K=0,1 for lanes 0-15
//   P23[s] = (-2z, x2+y2+z2) -> K=2,3 for lanes 16-31
// With B-operand (qx,qy,qz,1) the WMMA directly yields s^2 - 2 q.s.
// ---------------------------------------------------------------------------
__global__ __launch_bounds__(256) void rbf_prep_sensors(
    const float* __restrict__ sc, float2* __restrict__ P01,
    float2* __restrict__ P23) {
  int s = blockIdx.x * blockDim.x + threadIdx.x;
  if (s < kNS) {
    float x = sc[3 * s + 0], y = sc[3 * s + 1], z = sc[3 * s + 2];
    P01[s] = make_float2(-2.0f * x, -2.0f * y);
    P23[s] = make_float2(-2.0f * z, x * x + y * y + z * z);
  }
}

// Monotone float -> u32 mapping (order-preserving, handles negatives).
__device__ __forceinline__ u32 fmap(float f) {
  u32 u = __float_as_uint(f);
  return u ^ ((u32)((int)u >> 31) | 0x80000000u);
}
__device__ __forceinline__ float funmap(u32 m) {
  return __uint_as_float(m ^ ((u32)(~((int)m >> 31)) | 0x80000000u));
}

__device__ __forceinline__ u64 minu(u64 a, u64 b) { return a < b ? a : b; }
__device__ __forceinline__ u64 maxu(u64 a, u64 b) { return a < b ? b : a; }

// Branch-free sorted insert into ascending top-4 of packed (dist,idx) keys.
// Pure min/max-swap chain -> v_cmp_lt_u64 + v_cndmask, no EXEC churn.
__device__ __forceinline__ void insert4(u64 v, u64 k[4]) {
  u64 a;
  a = minu(k[0], v); v = maxu(k[0], v); k[0] = a;
  a = minu(k[1], v); v = maxu(k[1], v); k[1] = a;
  a = minu(k[2], v); v = maxu(k[2], v); k[2] = a;
  k[3] = minu(k[3], v);
}

__device__ __forceinline__ float4 fma4(float w, float4 v, float4 a) {
  a.x = fmaf(w, v.x, a.x); a.y = fmaf(w, v.y, a.y);
  a.z = fmaf(w, v.z, a.z); a.w = fmaf(w, v.w, a.w);
  return a;
}

__device__ __forceinline__ v8f wmma_tile(v2f a, v2f b) {
  v8f c = {};
  return __builtin_amdgcn_wmma_f32_16x16x4_f32(false, a, false, b, (short)0, c,
                                               false, false);
}

// ---------------------------------------------------------------------------
// One wave = 16 queries. WMMA D[m,n]: lanes 0-15 hold n=lane, sensors m=0..7
// (VGPR r), lanes 16-31 hold the same queries, sensors m=8..15.
// ---------------------------------------------------------------------------
__global__ __launch_bounds__(256) void rbf_knn_interp(
    const float* __restrict__ qc, const float2* __restrict__ P01,
    const float2* __restrict__ P23, const float* __restrict__ feat,
    const float* __restrict__ sigma_p, float* __restrict__ out) {
  const int lane   = threadIdx.x & 31;
  const int wave   = threadIdx.x >> 5;
  const int tileId = blockIdx.x * 8 + wave;          // 0..2047
  const int b      = tileId / kQTperB;
  const int qbase  = (tileId % kQTperB) * 16;
  const int l15    = lane & 15;
  const int n      = qbase + l15;                    // query owned by lane
  const bool hi    = lane >= 16;
  const int mbase  = hi ? 8 : 0;

  // B operand (queries): K rows (qx,qy) on lanes 0-15, (qz,1) on lanes 16-31.
  const float* qp = qc + ((size_t)b * kNQ + n) * 3;
  float qx = qp[0], qy = qp[1], qz = qp[2];
  float q2 = qx * qx + qy * qy + qz * qz;
  v2f bop;
  bop[0] = hi ? qz : qx;
  bop[1] = hi ? 1.0f : qy;

  const float2* __restrict__ Abase = hi ? P23 : P01;

  u64 k[4] = {~0ull, ~0ull, ~0ull, ~0ull};
  float thresh = 3.0e38f;

  // Software pipeline: keep two A-tiles in flight ahead of the WMMAs.
  float2 av0 = Abase[0 * 16 + l15];
  float2 av1 = Abase[1 * 16 + l15];

  for (int t = 0; t < kTiles; t += 2) {
    int t2 = (t + 2) & (kTiles - 1);
    int t3 = (t + 3) & (kTiles - 1);
    float2 nx0 = Abase[t2 * 16 + l15];
    float2 nx1 = Abase[t3 * 16 + l15];

    v2f a0; a0[0] = av0.x; a0[1] = av0.y;
    v2f a1; a1[0] = av1.x; a1[1] = av1.y;
    // d[r] = s2[m] - 2 q.s   (q2 is monotone-constant per lane; added later)
    v8f d0 = wmma_tile(a0, bop);
    v8f d1 = wmma_tile(a1, bop);

    float m0 = fminf(fminf(fminf(d0[0], d0[1]), fminf(d0[2], d0[3])),
                     fminf(fminf(d0[4], d0[5]), fminf(d0[6], d0[7])));
    float m1 = fminf(fminf(fminf(d1[0], d1[1]), fminf(d1[2], d1[3])),
                     fminf(fminf(d1[4], d1[5]), fminf(d1[6], d1[7])));
    if (fminf(m0, m1) < thresh) {                    // rare after warm-up
      int base0 = t * 16 + mbase;
      #pragma unroll
      for (int r = 0; r < 8; ++r)
        insert4(((u64)fmap(d0[r]) << 32) | (u32)(base0 + r), k);
      int base1 = base0 + 16;
      #pragma unroll
      for (int r = 0; r < 8; ++r)
        insert4(((u64)fmap(d1[r]) << 32) | (u32)(base1 + r), k);
      thresh = funmap((u32)(k[3] >> 32));
    }
    av0 = nx0;
    av1 = nx1;
  }

  // Merge with partner lane (same query, other 8 sensor rows per tile).
  // Snapshot first, then insert, so lockstep inserts can't cross-pollute.
  u64 ok[4];
  #pragma unroll
  for (int r = 0; r < 4; ++r) {
    int khi = __shfl_xor((int)(u32)(k[r] >> 32), 16, 32);
    int klo = __shfl_xor((int)(u32)k[r], 16, 32);
    ok[r] = ((u64)(u32)khi << 32) | (u32)klo;
  }
  #pragma unroll
  for (int r = 0; r < 4; ++r) insert4(ok[r], k);

  // RBF weights (restore the +q^2 term, clamp at 0 like the reference).
  float sg   = sigma_p[0];
  float sigp = log1pf(__expf(sg)) + 0.01f;           // softplus(sigma) + 0.01
  float inv  = 1.0f / (2.0f * sigp * sigp);
  float w[4];
  int   bidx[4];
  float wsum = 1e-5f;
  #pragma unroll
  for (int r = 0; r < 4; ++r) {
    bidx[r]  = (int)(u32)k[r];
    float d2 = fmaxf(funmap((u32)(k[r] >> 32)) + q2, 0.0f);
    w[r] = __expf(-d2 * inv);
    wsum += w[r];
  }
  float rws = 1.0f / wsum;
  #pragma unroll
  for (int r = 0; r < 4; ++r) w[r] *= rws;

  // Feature gather: one query at a time, whole wave covers 256 floats
  // (lane handles 8 contiguous floats = 2x float4). Fully coalesced.
  const float* fb = feat + (size_t)b * kNS * kF;
  float* ob = out + ((size_t)b * kNQ + qbase) * kF;
  const int c = lane * 2;                            // float4 column index
  for (int q = 0; q < 16; ++q) {
    float w0 = __shfl(w[0], q, 32), w1 = __shfl(w[1], q, 32);
    float w2 = __shfl(w[2], q, 32), w3 = __shfl(w[3], q, 32);
    int i0 = __shfl(bidx[0], q, 32), i1 = __shfl(bidx[1], q, 32);
    int i2 = __shfl(bidx[2], q, 32), i3 = __shfl(bidx[3], q, 32);

    const float4* r0 = (const float4*)(fb + (size_t)i0 * kF);
    const float4* r1 = (const float4*)(fb + (size_t)i1 * kF);
    const float4* r2 = (const float4*)(fb + (size_t)i2 * kF);
    const float4* r3 = (const float4*)(fb + (size_t)i3 * kF);

    float4 acc0 = make_float4(0.f, 0.f, 0.f, 0.f);
    float4 acc1 = make_float4(0.f, 0.f, 0.f, 0.f);
    acc0 = fma4(w0, r0[c], acc0);  acc1 = fma4(w0, r0[c + 1], acc1);
    acc0 = fma4(w1, r1[c], acc0);  acc1 = fma4(w1, r1[c + 1], acc1);
    acc0 = fma4(w2, r2[c], acc0);  acc1 = fma4(w2, r2[c + 1], acc1);
    acc0 = fma4(w3, r3[c], acc0);  acc1 = fma4(w3, r3[c + 1], acc1);

    float4* op = (float4*)(ob + (size_t)q * kF);
    op[c] = acc0;
    op[c + 1] = acc1;
  }
}

extern "C" void kernel_launch(void* const* d_in, const int* in_sizes, int n_in,
                              void* d_out, int out_size, void* d_ws, size_t ws_size,
                              hipStream_t stream) {
  const float* qc    = (const float*)d_in[0];   // (B, NQ, 3) f32
  const float* sc    = (const float*)d_in[1];   // (NS, 3)    f32
  const float* feat  = (const float*)d_in[2];   // (B, NS, F) f32
  const float* sigma = (const float*)d_in[3];   // scalar     f32
  // d_in[4] = k (== 4, hardcoded)
  float* out = (float*)d_out;                   // (B, NQ, F) f32

  float2* P01 = (float2*)d_ws;                  // 32 KB
  float2* P23 = P01 + kNS;                      // 32 KB

  rbf_prep_sensors<<<kNS / 256, 256, 0, stream>>>(sc, P01, P23);
  rbf_knn_interp<<<(kB * kQTperB) / 8, 256, 0, stream>>>(qc, P01, P23, feat,
                                                         sigma, out);
}